// LambdaLayer_61435212202589
// MI455X (gfx1250) — compile-verified
//
#include <hip/hip_runtime.h>
#include <hip/hip_bf16.h>

typedef __attribute__((ext_vector_type(2))) float v2f;
typedef __attribute__((ext_vector_type(8))) float v8f;

#if defined(__has_builtin)
#if __has_builtin(__builtin_amdgcn_global_load_async_to_lds_b128) && \
    __has_builtin(__builtin_amdgcn_s_wait_asynccnt)
#define HAVE_ASYNC_LDS 1
#endif
#endif

#ifdef HAVE_ASYNC_LDS
// match the diagnostic-reported prototype: (int __vector(4) __device__*, ...)
typedef int v4i __attribute__((vector_size(16)));
typedef __attribute__((address_space(1))) v4i* gas_v4i;   // global ptr to int4
typedef __attribute__((address_space(3))) v4i* las_v4i;   // LDS ptr to int4

__device__ __forceinline__ void async_copy_b128(const float* gsrc, float* ldst) {
    __builtin_amdgcn_global_load_async_to_lds_b128(
        (gas_v4i)(float*)(gsrc), (las_v4i)(ldst), 0, 0);
}
#endif

__device__ __forceinline__ v8f wmma4(v2f a, v2f b, v8f c) {
    // V_WMMA_F32_16X16X4_F32 : D(16x16,f32) = A(16x4) x B(4x16) + C
    return __builtin_amdgcn_wmma_f32_16x16x4_f32(
        false, a, false, b, (short)0, c, false, false);
}

#define BN_EPS 1e-3f

// ---------------------------------------------------------------------------
// Kernel 1: fused projection GEMM.  C(1024 x 1536) = X(1024x1024) @ [Wq|Wk|Wv]
// columns [0,1024) -> Q (+BN), [1024,1280) -> K, [1280,1536) -> V (+BN).
// Block: 256 thr (8 waves), tile 64(M) x 64(N), K staged in LDS 16 at a time.
// B tile staging uses GLOBAL_LOAD_ASYNC_TO_LDS_B128 when available.
// ---------------------------------------------------------------------------
__global__ __launch_bounds__(256) void proj_kernel(
    const float* __restrict__ X,
    const float* __restrict__ Wq, const float* __restrict__ Wk,
    const float* __restrict__ Wv,
    const float* __restrict__ qg, const float* __restrict__ qb,
    const float* __restrict__ qm, const float* __restrict__ qvar,
    const float* __restrict__ vg, const float* __restrict__ vb,
    const float* __restrict__ vm, const float* __restrict__ vvar,
    float* __restrict__ Q, float* __restrict__ K, float* __restrict__ V)
{
    __shared__ float As[16 * 64];   // [k][m]
    __shared__ float Bs[16 * 64];   // [k][n]

    const int tid  = threadIdx.x;
    const int lane = tid & 31;
    const int wid  = tid >> 5;
    const int l16  = lane & 15;
    const int lhi  = lane >> 4;
    const int wm   = wid & 3;       // 4 wave rows  -> 64 M
    const int wn   = wid >> 2;      // 2 wave cols  -> 2 x 32 N
    const int bm   = blockIdx.x * 64;
    const int bn   = blockIdx.y * 64;   // global column in [0,1536)

    // pick weight segment for this 64-wide column block (uniform per block)
    const float* Bsrc; int ldb, coln0;
    if (bn < 1024)      { Bsrc = Wq; ldb = 1024; coln0 = bn; }
    else if (bn < 1280) { Bsrc = Wk; ldb = 256;  coln0 = bn - 1024; }
    else                { Bsrc = Wv; ldb = 256;  coln0 = bn - 1280; }

    const int ar = tid >> 2, ak = (tid & 3) * 4;   // A stage: 64 rows x 16 k
    const int bk = tid >> 4, bc = (tid & 15) * 4;  // B stage: 16 k x 64 n

    v8f acc[2] = {};

    for (int k0 = 0; k0 < 1024; k0 += 16) {
        __syncthreads();
        // prefetch next A tile into cache while we stage this one
        if (k0 + 16 < 1024)
            __builtin_prefetch(&X[(bm + ar) * 1024 + k0 + 16 + ak], 0, 0);
#ifdef HAVE_ASYNC_LDS
        // B tile is a direct byte copy: use the async to-LDS path (ASYNCcnt)
        async_copy_b128(&Bsrc[(k0 + bk) * ldb + coln0 + bc], &Bs[bk * 64 + bc]);
#else
        float4 bv = *(const float4*)&Bsrc[(k0 + bk) * ldb + coln0 + bc];
        *(float4*)&Bs[bk * 64 + bc] = bv;
#endif
        float4 av = *(const float4*)&X[(bm + ar) * 1024 + k0 + ak];
        As[(ak + 0) * 64 + ar] = av.x;
        As[(ak + 1) * 64 + ar] = av.y;
        As[(ak + 2) * 64 + ar] = av.z;
        As[(ak + 3) * 64 + ar] = av.w;
#ifdef HAVE_ASYNC_LDS
        __builtin_amdgcn_s_wait_asynccnt(0);
#endif
        __syncthreads();

#pragma unroll
        for (int kk = 0; kk < 4; ++kk) {
            const int kl = kk * 4 + lhi * 2;
            v2f a;
            a[0] = As[(kl + 0) * 64 + wm * 16 + l16];
            a[1] = As[(kl + 1) * 64 + wm * 16 + l16];
#pragma unroll
            for (int nt = 0; nt < 2; ++nt) {
                v2f bf;
                bf[0] = Bs[(kl + 0) * 64 + wn * 32 + nt * 16 + l16];
                bf[1] = Bs[(kl + 1) * 64 + wn * 32 + nt * 16 + l16];
                acc[nt] = wmma4(a, bf, acc[nt]);
            }
        }
    }

    // epilogue: per-column BN affine, route to Q/K/V
#pragma unroll
    for (int nt = 0; nt < 2; ++nt) {
        const int gcol = bn + wn * 32 + nt * 16 + l16;
        float scale = 1.f, bias = 0.f;
        float* Out; int ld, d;
        if (gcol < 1024) {
            d = gcol; Out = Q; ld = 1024;
            float s = qg[d] * rsqrtf(qvar[d] + BN_EPS);
            scale = s; bias = qb[d] - qm[d] * s;
        } else if (gcol < 1280) {
            d = gcol - 1024; Out = K; ld = 256;
        } else {
            d = gcol - 1280; Out = V; ld = 256;
            float s = vg[d] * rsqrtf(vvar[d] + BN_EPS);
            scale = s; bias = vb[d] - vm[d] * s;
        }
#pragma unroll
        for (int r = 0; r < 8; ++r) {
            const int row = bm + wm * 16 + r + 8 * lhi;
            Out[row * ld + d] = acc[nt][r] * scale + bias;
        }
    }
}

// ---------------------------------------------------------------------------
// Kernel 2: softmax over each 256-wide row of K (1024 rows). 1 wave / row.
// ---------------------------------------------------------------------------
__global__ __launch_bounds__(256) void softmax_kernel(float* __restrict__ Kb)
{
    const int row  = blockIdx.x * 8 + (threadIdx.x >> 5);
    const int lane = threadIdx.x & 31;
    float* p = Kb + row * 256;
    float4 x0 = *(const float4*)&p[lane * 8];
    float4 x1 = *(const float4*)&p[lane * 8 + 4];
    float mx = fmaxf(fmaxf(fmaxf(x0.x, x0.y), fmaxf(x0.z, x0.w)),
                     fmaxf(fmaxf(x1.x, x1.y), fmaxf(x1.z, x1.w)));
#pragma unroll
    for (int off = 16; off > 0; off >>= 1) mx = fmaxf(mx, __shfl_xor(mx, off, 32));
    x0.x = __expf(x0.x - mx); x0.y = __expf(x0.y - mx);
    x0.z = __expf(x0.z - mx); x0.w = __expf(x0.w - mx);
    x1.x = __expf(x1.x - mx); x1.y = __expf(x1.y - mx);
    x1.z = __expf(x1.z - mx); x1.w = __expf(x1.w - mx);
    float s = x0.x + x0.y + x0.z + x0.w + x1.x + x1.y + x1.z + x1.w;
#pragma unroll
    for (int off = 16; off > 0; off >>= 1) s += __shfl_xor(s, off, 32);
    const float inv = 1.f / s;
    x0.x *= inv; x0.y *= inv; x0.z *= inv; x0.w *= inv;
    x1.x *= inv; x1.y *= inv; x1.z *= inv; x1.w *= inv;
    *(float4*)&p[lane * 8]     = x0;
    *(float4*)&p[lane * 8 + 4] = x1;
}

// ---------------------------------------------------------------------------
// Kernel 3: Lc[b](256x256) = Ksm[b] @ V[b]^T   (NT GEMM, contraction over m)
// grid (mtile=4, ntile=4, b=4), block tile 64x64.
// ---------------------------------------------------------------------------
__global__ __launch_bounds__(256) void lc_kernel(
    const float* __restrict__ Ksm, const float* __restrict__ Vb,
    float* __restrict__ Lc)
{
    __shared__ float As[16 * 64];   // [m][kk]
    __shared__ float Bs[16 * 64];   // [m][vv]
    const int tid  = threadIdx.x;
    const int lane = tid & 31, wid = tid >> 5;
    const int l16  = lane & 15, lhi = lane >> 4;
    const int wm   = wid & 3, wn = wid >> 2;
    const int b    = blockIdx.z;
    const int bm   = blockIdx.x * 64;   // kk base
    const int bn   = blockIdx.y * 64;   // vv base
    const float* A = Ksm + b * 65536;   // [kk][m]
    const float* B = Vb  + b * 65536;   // [vv][m]

    const int ar = tid >> 2, ak = (tid & 3) * 4;
    v8f acc[2] = {};

    for (int m0 = 0; m0 < 256; m0 += 16) {
        __syncthreads();
        float4 av = *(const float4*)&A[(bm + ar) * 256 + m0 + ak];
        As[(ak + 0) * 64 + ar] = av.x;
        As[(ak + 1) * 64 + ar] = av.y;
        As[(ak + 2) * 64 + ar] = av.z;
        As[(ak + 3) * 64 + ar] = av.w;
        float4 bv = *(const float4*)&B[(bn + ar) * 256 + m0 + ak];
        Bs[(ak + 0) * 64 + ar] = bv.x;
        Bs[(ak + 1) * 64 + ar] = bv.y;
        Bs[(ak + 2) * 64 + ar] = bv.z;
        Bs[(ak + 3) * 64 + ar] = bv.w;
        __syncthreads();

#pragma unroll
        for (int kk = 0; kk < 4; ++kk) {
            const int kl = kk * 4 + lhi * 2;
            v2f a;
            a[0] = As[(kl + 0) * 64 + wm * 16 + l16];
            a[1] = As[(kl + 1) * 64 + wm * 16 + l16];
#pragma unroll
            for (int nt = 0; nt < 2; ++nt) {
                v2f bf;
                bf[0] = Bs[(kl + 0) * 64 + wn * 32 + nt * 16 + l16];
                bf[1] = Bs[(kl + 1) * 64 + wn * 32 + nt * 16 + l16];
                acc[nt] = wmma4(a, bf, acc[nt]);
            }
        }
    }
#pragma unroll
    for (int nt = 0; nt < 2; ++nt) {
        const int col = bn + wn * 32 + nt * 16 + l16;
#pragma unroll
        for (int r = 0; r < 8; ++r) {
            const int row = bm + wm * 16 + r + 8 * lhi;
            Lc[b * 65536 + row * 256 + col] = acc[nt][r];
        }
    }
}

// ---------------------------------------------------------------------------
// Kernel 4: positional path, kk contracted first (64x fewer FLOPs than ref
// ordering).  For each n: T[n](16x256) = Qstack_n(16x256) @ Rel_n^T,
// rows = (b*4+h), cols = m.  Rel_n[m][kk] = E[(di(m)*31+dj(m))*256 + kk].
// 1 block per n; A staged in LDS; rel rows gathered from L2-resident table.
// ---------------------------------------------------------------------------
__global__ __launch_bounds__(256) void t_kernel(
    const float* __restrict__ Qb, const float* __restrict__ E,
    float* __restrict__ T)
{
    __shared__ float As[256 * 16];  // [kk][row]
    const int nn  = blockIdx.x;
    const int tid = threadIdx.x;

    // gather A: As[kk*16 + r] = Q[b=r>>2, h=r&3, kk, nn]
    {
        const int r = tid >> 4;
        const int base = (r >> 2) * 262144 + (r & 3) * 65536 + nn;
#pragma unroll
        for (int i = 0; i < 16; ++i) {
            const int kk = (tid & 15) + i * 16;
            As[kk * 16 + r] = Qb[base + kk * 256];
        }
    }
    __syncthreads();

    const int lane = tid & 31, wid = tid >> 5;
    const int l16  = lane & 15, lhi = lane >> 4;
    const int i_n  = nn >> 4, j_n = nn & 15;

    int eidx[2];
#pragma unroll
    for (int nt = 0; nt < 2; ++nt) {
        const int m  = wid * 32 + nt * 16 + l16;
        const int di = (m >> 4) - i_n + 15;
        const int dj = (m & 15) - j_n + 15;
        eidx[nt] = (di * 31 + dj) * 256;
    }

    v8f acc[2] = {};
    for (int k0 = 0; k0 < 256; k0 += 4) {
        const int kl = k0 + lhi * 2;
        v2f a;
        a[0] = As[(kl + 0) * 16 + l16];
        a[1] = As[(kl + 1) * 16 + l16];
#pragma unroll
        for (int nt = 0; nt < 2; ++nt) {
            const float2 ev = *(const float2*)&E[eidx[nt] + kl];
            v2f bf; bf[0] = ev.x; bf[1] = ev.y;
            acc[nt] = wmma4(a, bf, acc[nt]);
        }
    }

#pragma unroll
    for (int nt = 0; nt < 2; ++nt) {
        const int m = wid * 32 + nt * 16 + l16;
#pragma unroll
        for (int r = 0; r < 8; ++r) {
            const int row = r + 8 * lhi;
            T[nn * 4096 + row * 256 + m] = acc[nt][r];
        }
    }
}

// ---------------------------------------------------------------------------
// Kernel 5: fused output.  Per (b, group g of 4 n's):
//   Y(16x256) = Acat(16x512) @ Bcat(512x256)
// Acat rows = (nn_off*4 + h); k<256: T (positional, contract m),
// k>=256: Q columns (content, contract kk).  Bcat: k<256: V^T; k>=256: Lc.
// Writes Yc+Yp straight to d_out (no read-modify-write => replay-safe).
// ---------------------------------------------------------------------------
__global__ __launch_bounds__(256) void y_kernel(
    const float* __restrict__ Qb, const float* __restrict__ Vb,
    const float* __restrict__ Lc, const float* __restrict__ T,
    float* __restrict__ Y)
{
    __shared__ float As[512 * 16];  // [k][row]  (32 KB)
    const int g = blockIdx.x;       // n group: nn = g*4 + nn_off
    const int b = blockIdx.y;
    const int tid = threadIdx.x;

    {
        const int r = tid >> 4;
        const int nn = g * 4 + (r >> 2);
        const int h  = r & 3;
        const float* trow = T + nn * 4096 + (b * 4 + h) * 256;
        const int qbase = b * 262144 + h * 65536 + nn;
#pragma unroll
        for (int i = 0; i < 32; ++i) {
            const int k = (tid & 15) * 32 + i;
            As[k * 16 + r] = (k < 256) ? trow[k] : Qb[qbase + (k - 256) * 256];
        }
    }
    __syncthreads();

    const int lane = tid & 31, wid = tid >> 5;
    const int l16  = lane & 15, lhi = lane >> 4;

    v8f acc[2] = {};
    for (int k0 = 0; k0 < 512; k0 += 4) {
        const int kl = k0 + lhi * 2;
        v2f a;
        a[0] = As[(kl + 0) * 16 + l16];
        a[1] = As[(kl + 1) * 16 + l16];
#pragma unroll
        for (int nt = 0; nt < 2; ++nt) {
            const int n = wid * 32 + nt * 16 + l16;   // vv
            v2f bf;
            if (kl < 256) {  // V^T : B[m][vv] = V[vv][m]
                const float2 ev = *(const float2*)&Vb[b * 65536 + n * 256 + kl];
                bf[0] = ev.x; bf[1] = ev.y;
            } else {         // Lc  : B[kk][vv]
                bf[0] = Lc[b * 65536 + (kl - 256) * 256 + n];
                bf[1] = Lc[b * 65536 + (kl - 255) * 256 + n];
            }
            acc[nt] = wmma4(a, bf, acc[nt]);
        }
    }

#pragma unroll
    for (int nt = 0; nt < 2; ++nt) {
        const int vv = wid * 32 + nt * 16 + l16;
#pragma unroll
        for (int r = 0; r < 8; ++r) {
            const int row = r + 8 * lhi;           // = nn_off*4 + h
            const int nn  = g * 4 + (row >> 2);
            const int h   = row & 3;
            Y[b * 262144 + nn * 1024 + h * 256 + vv] = acc[nt][r];
        }
    }
}

// ---------------------------------------------------------------------------
extern "C" void kernel_launch(void* const* d_in, const int* in_sizes, int n_in,
                              void* d_out, int out_size, void* d_ws, size_t ws_size,
                              hipStream_t stream) {
    const float* x    = (const float*)d_in[0];
    const float* Wq   = (const float*)d_in[1];
    const float* Wk   = (const float*)d_in[2];
    const float* Wv   = (const float*)d_in[3];
    const float* qg   = (const float*)d_in[4];
    const float* qb   = (const float*)d_in[5];
    const float* qm   = (const float*)d_in[6];
    const float* qvar = (const float*)d_in[7];
    const float* vg   = (const float*)d_in[8];
    const float* vb   = (const float*)d_in[9];
    const float* vm   = (const float*)d_in[10];
    const float* vvar = (const float*)d_in[11];
    const float* E    = (const float*)d_in[12];
    float* Y = (float*)d_out;

    float* Qw  = (float*)d_ws;          // 1M floats
    float* Kw  = Qw  + (1 << 20);       // 256K
    float* Vw  = Kw  + (1 << 18);       // 256K
    float* Lcw = Vw  + (1 << 18);       // 256K
    float* Tw  = Lcw + (1 << 18);       // 1M

    proj_kernel<<<dim3(16, 24), 256, 0, stream>>>(
        x, Wq, Wk, Wv, qg, qb, qm, qvar, vg, vb, vm, vvar, Qw, Kw, Vw);
    softmax_kernel<<<128, 256, 0, stream>>>(Kw);
    lc_kernel<<<dim3(4, 4, 4), 256, 0, stream>>>(Kw, Vw, Lcw);
    t_kernel<<<256, 256, 0, stream>>>(Qw, E, Tw);
    y_kernel<<<dim3(64, 4), 256, 0, stream>>>(Qw, Vw, Lcw, Tw, Y);
}